// GCNconv_90615220011124
// MI455X (gfx1250) — compile-verified
//
#include <hip/hip_runtime.h>
#include <hip/hip_bf16.h>

typedef __attribute__((ext_vector_type(2))) float v2f;
typedef __attribute__((ext_vector_type(8))) float v8f;

#define GCN_N_NODES 100000
#define GCN_N_EDGES 1000000
#define GCN_D 64

// ---------------------------------------------------------------- deg = 0
__global__ void gcn_zero_deg(float* __restrict__ deg) {
    int i = blockIdx.x * blockDim.x + threadIdx.x;
    if (i < GCN_N_NODES) deg[i] = 0.0f;
}

// ---------------------------------------------------------------- deg[src] += 1
__global__ void gcn_degree(const int* __restrict__ edge_index, float* __restrict__ deg) {
    int e = blockIdx.x * blockDim.x + threadIdx.x;
    if (e < GCN_N_EDGES) {
        atomicAdd(&deg[edge_index[e]], 1.0f);  // edge_index[0][e] = src
    }
}

// ---------------------------------------------------------------- x_norm = x * rsqrt(deg+1); agg = x_norm (self loop)
__global__ void gcn_normalize(const float* __restrict__ x, const float* __restrict__ deg,
                              float* __restrict__ x_norm, float* __restrict__ agg) {
    int i = blockIdx.x * blockDim.x + threadIdx.x;          // one float4 per thread
    if (i >= GCN_N_NODES * (GCN_D / 4)) return;
    int n = i / (GCN_D / 4);
    float s = rsqrtf(deg[n] + 1.0f);
    float4 v = ((const float4*)x)[i];
    v.x *= s; v.y *= s; v.z *= s; v.w *= s;
    ((float4*)x_norm)[i] = v;
    ((float4*)agg)[i]    = v;
}

// ---------------------------------------------------------------- agg[dst] += x_norm[src]  (64 lanes per edge)
__global__ void gcn_scatter(const int* __restrict__ edge_index,
                            const float* __restrict__ x_norm, float* __restrict__ agg) {
    int t = blockIdx.x * blockDim.x + threadIdx.x;
    int e = t >> 6;            // edge id
    int f = t & 63;            // feature id
    if (e >= GCN_N_EDGES) return;
    int s = edge_index[e];                  // src
    int d = edge_index[GCN_N_EDGES + e];    // dst
    atomicAdd(&agg[d * GCN_D + f], x_norm[s * GCN_D + f]);
}

// ---------------------------------------------------------------- out = agg @ W^T + b  via V_WMMA_F32_16X16X4_F32
// One wave32 per 16x16 output tile. K=64 -> 16 chained WMMAs.
// A layout (16x4 f32): lane holds row (lane&15), k-pair base kb=(lane>>4)*2  -> {A[r][kb],A[r][kb+1]}
// B layout (4x16 f32): lane holds col (lane&15), same kb                    -> {B[kb][c],B[kb+1][c]}
//   with B[k][o] = W[o][k]  (W is [64,64] row-major, out = agg @ W^T)
// C/D (16x16 f32, 8 VGPRs): lanes 0-15 -> M=r, lanes 16-31 -> M=8+r; N=lane&15
__global__ void gcn_gemm_wmma(const float* __restrict__ agg, const float* __restrict__ W,
                              const float* __restrict__ bias, float* __restrict__ out) {
    const int lane = threadIdx.x & 31;
    const int wave = threadIdx.x >> 5;                 // wave32 (gfx1250)
    const int tile = blockIdx.x * 8 + wave;            // 8 waves per 256-thread block
    const int TILES_N = GCN_D / 16;                    // 4
    const int n_tiles = (GCN_N_NODES / 16) * TILES_N;  // 25000
    if (tile >= n_tiles) return;                       // uniform per wave -> EXEC all ones inside

    const int tm = tile / TILES_N;
    const int tn = tile % TILES_N;
    const int m0 = tm * 16;
    const int o0 = tn * 16;

    const int row = lane & 15;
    const int col = lane & 15;
    const int kb  = (lane >> 4) << 1;                  // 0 or 2

    // Preload B operands for all 16 k-steps (W rows are contiguous in k).
    v2f breg[16];
    const float* wrow = W + (o0 + col) * GCN_D + kb;
#pragma unroll
    for (int t = 0; t < 16; ++t) {
        breg[t] = *(const v2f*)(wrow + 4 * t);         // {W[o][4t+kb], W[o][4t+kb+1]}
    }

    // Fold bias into the accumulator.
    const float bv = bias[o0 + col];
    v8f c;
#pragma unroll
    for (int r = 0; r < 8; ++r) c[r] = bv;

    const float* arow = agg + (m0 + row) * GCN_D + kb;
#pragma unroll
    for (int t = 0; t < 16; ++t) {
        v2f a = *(const v2f*)(arow + 4 * t);           // {agg[m][4t+kb], agg[m][4t+kb+1]}
        // (neg_a, A, neg_b, B, c_mod, C, reuse_a, reuse_b)
        c = __builtin_amdgcn_wmma_f32_16x16x4_f32(
                false, a, false, breg[t], (short)0, c, false, false);
    }

    // Store: lanes 0-15 cover rows m0..m0+7, lanes 16-31 rows m0+8..m0+15.
    float* orow = out + (m0 + ((lane >> 4) << 3)) * GCN_D + o0 + col;
#pragma unroll
    for (int r = 0; r < 8; ++r) {
        orow[r * GCN_D] = c[r];
    }
}

extern "C" void kernel_launch(void* const* d_in, const int* in_sizes, int n_in,
                              void* d_out, int out_size, void* d_ws, size_t ws_size,
                              hipStream_t stream) {
    const float* x          = (const float*)d_in[0];   // [100000, 64]
    const int*   edge_index = (const int*)d_in[1];     // [2, 1000000] flattened
    const float* W          = (const float*)d_in[2];   // [64, 64]
    const float* b          = (const float*)d_in[3];   // [64]
    float*       out        = (float*)d_out;           // [100000, 64]

    // Workspace layout (floats): deg[N] | x_norm[N*64] | agg[N*64]  (~52 MB)
    float* deg    = (float*)d_ws;
    float* x_norm = deg + GCN_N_NODES;
    float* agg    = x_norm + (size_t)GCN_N_NODES * GCN_D;

    gcn_zero_deg<<<(GCN_N_NODES + 255) / 256, 256, 0, stream>>>(deg);
    gcn_degree<<<(GCN_N_EDGES + 255) / 256, 256, 0, stream>>>(edge_index, deg);
    gcn_normalize<<<(GCN_N_NODES * (GCN_D / 4) + 255) / 256, 256, 0, stream>>>(x, deg, x_norm, agg);
    gcn_scatter<<<(GCN_N_EDGES * GCN_D) / 256, 256, 0, stream>>>(edge_index, x_norm, agg);

    const int n_tiles = (GCN_N_NODES / 16) * (GCN_D / 16);  // 25000 tiles, 8 waves/block
    gcn_gemm_wmma<<<(n_tiles + 7) / 8, 256, 0, stream>>>(agg, W, b, out);
}